// AUROCSurrogateLoss_90666759619072
// MI455X (gfx1250) — compile-verified
//
#include <hip/hip_runtime.h>

typedef __attribute__((ext_vector_type(2))) float v2f;
typedef __attribute__((ext_vector_type(8))) float v8f;

#define ROWS_PER_BLOCK 1024
#define COLS_PER_BLOCK 512   // 8 waves x 64 columns
#define THREADS 256
#define ROW_PAD 8            // prefetch overrun space (never consumed)

__device__ __forceinline__ float sigmoidf(float x) {
    return 1.0f / (1.0f + __expf(-x));
}

// ws layout: ws[0] = float total, ((int*)ws)[1] = P, ((int*)ws)[2] = Q
__global__ void auroc_init_ws(float* ws) {
    if (threadIdx.x == 0) {
        ws[0] = 0.0f;
        ((int*)ws)[1] = 0;
        ((int*)ws)[2] = 0;
    }
}

__global__ void auroc_count(const int* __restrict__ y_true, int n,
                            int* __restrict__ gP, int* __restrict__ gQ) {
    __shared__ int sP, sQ;
    const int tid = threadIdx.x;
    if (tid == 0) { sP = 0; sQ = 0; }
    __syncthreads();
    const int i = blockIdx.x * blockDim.x + tid;
    int p = 0, q = 0;
    if (i < n) {
        const int t = y_true[i];
        p = (t == 1);
        q = (t == 0);
    }
    unsigned long long mp = __ballot(p);
    unsigned long long mq = __ballot(q);
    if ((tid & 31) == 0) {
        atomicAdd(&sP, (int)__popcll(mp));
        atomicAdd(&sQ, (int)__popcll(mq));
    }
    __syncthreads();
    if (tid == 0) {
        atomicAdd(gP, sP);
        atomicAdd(gQ, sQ);
    }
}

// Pairwise hinge sum. Rows = potential positives (c_i = 1-s_i or -1e30),
// columns = potential negatives (v_j = s_j or -1e30); the sentinel drives
// invalid pairs to exactly 0 through max(.,0), so the loop is mask-free.
// Two rows' hinges are pre-accumulated in (dual-issued) VALU, then reduced
// on the matrix pipe with V_WMMA_F32_16X16X4_F32 (A = all-ones):
// sum(D) == 16 * sum(all B fed in), independent of B's lane layout.
__global__ void __launch_bounds__(THREADS)
auroc_pairs(const float* __restrict__ y_pred, const int* __restrict__ y_true,
            float* __restrict__ total, int n) {
    __shared__ __align__(16) float rowc[ROWS_PER_BLOCK + ROW_PAD];

    const int tid     = threadIdx.x;
    const int rowBase = blockIdx.y * ROWS_PER_BLOCK;
    const int colBase = blockIdx.x * COLS_PER_BLOCK;

    // Stage transformed row values into LDS (+pad for prefetch overrun).
    for (int r = tid; r < ROWS_PER_BLOCK + ROW_PAD; r += THREADS) {
        int i = rowBase + r;
        i = (i < n) ? i : (n - 1);
        const float s = sigmoidf(y_pred[i]);
        rowc[r] = (y_true[i] == 1) ? (1.0f - s) : -1.0e30f;
    }
    __syncthreads();

    const int wave = tid >> 5;
    const int lane = tid & 31;
    int j0 = colBase + wave * 64 + lane;
    int j1 = j0 + 32;
    j0 = (j0 < n) ? j0 : (n - 1);
    j1 = (j1 < n) ? j1 : (n - 1);
    const float v0 = (y_true[j0] == 0) ? sigmoidf(y_pred[j0]) : -1.0e30f;
    const float v1 = (y_true[j1] == 0) ? sigmoidf(y_pred[j1]) : -1.0e30f;

    v2f a;
    a[0] = 1.0f;
    a[1] = 1.0f;
    v8f acc0 = {};
    v8f acc1 = {};

    // Software-pipelined over 8 rows/iteration: prefetch next two float4s of
    // row values while the current ones feed the VALU + matrix pipe.
    // EXEC is all-ones for every thread here (no divergence) as WMMA requires.
    const float4* rowv = (const float4*)rowc;
    float4 cur0 = rowv[0];
    float4 cur1 = rowv[1];

    for (int q = 0; q < ROWS_PER_BLOCK / 8; ++q) {
        const float4 nxt0 = rowv[2 * q + 2];
        const float4 nxt1 = rowv[2 * q + 3];

        v2f b;
        // rows cur0.x, cur0.y (2 rows pre-added per WMMA feed)
        b[0] = fmaxf(cur0.x + v0, 0.0f) + fmaxf(cur0.y + v0, 0.0f);
        b[1] = fmaxf(cur0.x + v1, 0.0f) + fmaxf(cur0.y + v1, 0.0f);
        acc0 = __builtin_amdgcn_wmma_f32_16x16x4_f32(
            false, a, false, b, (short)0, acc0, false, false);
        // rows cur0.z, cur0.w
        b[0] = fmaxf(cur0.z + v0, 0.0f) + fmaxf(cur0.w + v0, 0.0f);
        b[1] = fmaxf(cur0.z + v1, 0.0f) + fmaxf(cur0.w + v1, 0.0f);
        acc1 = __builtin_amdgcn_wmma_f32_16x16x4_f32(
            false, a, false, b, (short)0, acc1, false, false);
        // rows cur1.x, cur1.y
        b[0] = fmaxf(cur1.x + v0, 0.0f) + fmaxf(cur1.y + v0, 0.0f);
        b[1] = fmaxf(cur1.x + v1, 0.0f) + fmaxf(cur1.y + v1, 0.0f);
        acc0 = __builtin_amdgcn_wmma_f32_16x16x4_f32(
            false, a, false, b, (short)0, acc0, false, false);
        // rows cur1.z, cur1.w
        b[0] = fmaxf(cur1.z + v0, 0.0f) + fmaxf(cur1.w + v0, 0.0f);
        b[1] = fmaxf(cur1.z + v1, 0.0f) + fmaxf(cur1.w + v1, 0.0f);
        acc1 = __builtin_amdgcn_wmma_f32_16x16x4_f32(
            false, a, false, b, (short)0, acc1, false, false);

        cur0 = nxt0;
        cur1 = nxt1;
    }

    // sum(D over all lanes/VGPRs) = 16 * sum(hinges this wave consumed)
    float lsum = 0.0f;
    #pragma unroll
    for (int k = 0; k < 8; ++k) lsum += acc0[k] + acc1[k];
    #pragma unroll
    for (int off = 16; off > 0; off >>= 1)
        lsum += __shfl_down(lsum, off, 32);
    if (lane == 0)
        atomicAdd(total, lsum * (1.0f / 16.0f));
}

__global__ void auroc_finalize(const float* __restrict__ ws,
                               float* __restrict__ out) {
    const float total = ws[0];
    const float P = (float)((const int*)ws)[1];
    const float Q = (float)((const int*)ws)[2];
    const float denom = P * Q;
    out[0] = (denom > 0.0f) ? (total / fmaxf(denom, 1.0f)) : 0.0f;
}

extern "C" void kernel_launch(void* const* d_in, const int* in_sizes, int n_in,
                              void* d_out, int out_size, void* d_ws, size_t ws_size,
                              hipStream_t stream) {
    const float* y_pred = (const float*)d_in[0];
    const int*   y_true = (const int*)d_in[1];
    float* ws = (float*)d_ws;
    const int n = in_sizes[0];   // 16384

    auroc_init_ws<<<1, 1, 0, stream>>>(ws);
    auroc_count<<<(n + THREADS - 1) / THREADS, THREADS, 0, stream>>>(
        y_true, n, (int*)ws + 1, (int*)ws + 2);

    dim3 grid((n + COLS_PER_BLOCK - 1) / COLS_PER_BLOCK,
              (n + ROWS_PER_BLOCK - 1) / ROWS_PER_BLOCK);  // 32 x 16 = 512 blocks
    auroc_pairs<<<grid, THREADS, 0, stream>>>(y_pred, y_true, ws, n);

    auroc_finalize<<<1, 1, 0, stream>>>(ws, (float*)d_out);
}